// Voxelization_19739669693073
// MI455X (gfx1250) — compile-verified
//
#include <hip/hip_runtime.h>
#include <hip/hip_bf16.h>
#include <limits.h>
#include <stdint.h>

// ---------------- problem constants (from reference) ----------------
#define GX 1024
#define GY 1024
#define GZ 40
#define GYZ (GY * GZ)              // 40960
#define MAX_POINTS 10
#define MAX_VOXELS 120000
#define NFEAT 5

// output layout (floats, concatenated in reference return order)
#define VOX_ELEMS   (MAX_VOXELS * MAX_POINTS * NFEAT)   // 6,000,000
#define COOR_ELEMS  (MAX_VOXELS * 3)                    //   360,000
#define NPV_ELEMS   (MAX_VOXELS)                        //   120,000
#define VOXNUM_OFF  (VOX_ELEMS + COOR_ELEMS + NPV_ELEMS)

// hash map: open addressing, 4M slots for <=1.2M keys (load factor < 0.3)
#define HCAP  (1u << 22)
#define HMASK (HCAP - 1u)
#define KEMPTY (-1)

__device__ __forceinline__ unsigned hslot(int lin) {
  return ((unsigned)lin * 2654435761u) & HMASK;
}

__device__ __forceinline__ int hfind(const int* __restrict__ keys, int lin) {
  unsigned s = hslot(lin);
  while (keys[s] != lin) s = (s + 1u) & HMASK;
  return (int)s;
}

__device__ __forceinline__ int hinsert(int* keys, int lin) {
  unsigned s = hslot(lin);
  for (;;) {
    int k = atomicCAS(&keys[s], KEMPTY, lin);
    if (k == KEMPTY || k == lin) return (int)s;
    s = (s + 1u) & HMASK;
  }
}

// ---------------- kernels ----------------

__global__ void k_init_hash(int* keys, int* vals) {
  unsigned i = blockIdx.x * blockDim.x + threadIdx.x;
  if (i < HCAP) { keys[i] = KEMPTY; vals[i] = INT_MAX; }
}

__global__ void k_init_vstate(int* cnt, int* thr, int* cmin) {
  int r = blockIdx.x * blockDim.x + threadIdx.x;
  if (r < MAX_VOXELS) { cnt[r] = 0; thr[r] = -1; cmin[r] = INT_MAX; }
}

__global__ void k_zero_out(float* out, int n) {
  int i = blockIdx.x * blockDim.x + threadIdx.x;
  if (i < n) out[i] = 0.0f;
}

// per-point linear voxel id (or -1) AND first-index atomicMin, fused
__global__ void k_lin_first(const float* __restrict__ pts, int* __restrict__ lin,
                            int* keys, int* vals, int n) {
  int i = blockIdx.x * blockDim.x + threadIdx.x;
  if (i >= n) return;
  // gfx1250 prefetch of the point stream ahead of this wavefront
  __builtin_prefetch(pts + (size_t)i * NFEAT + 4096, 0, 0);
  float x = pts[(size_t)i * NFEAT + 0];
  float y = pts[(size_t)i * NFEAT + 1];
  float z = pts[(size_t)i * NFEAT + 2];
  int ix = (int)floorf((x - (-51.2f)) / 0.1f);
  int iy = (int)floorf((y - (-51.2f)) / 0.1f);
  int iz = (int)floorf((z - (-5.0f)) / 0.2f);
  bool ok = (ix >= 0) & (ix < GX) & (iy >= 0) & (iy < GY) & (iz >= 0) & (iz < GZ);
  int l = ok ? (ix * GYZ + iy * GZ + iz) : -1;
  lin[i] = l;
  if (l >= 0) {
    int s = hinsert(keys, l);
    atomicMin(&vals[s], i);
  }
}

// is_first flag per point
__global__ void k_flags(const int* __restrict__ lin, const int* __restrict__ keys,
                        const int* __restrict__ vals, int* __restrict__ flg, int n) {
  int i = blockIdx.x * blockDim.x + threadIdx.x;
  if (i >= n) return;
  int l = lin[i];
  int f = 0;
  if (l >= 0) f = (vals[hfind(keys, l)] == i) ? 1 : 0;
  flg[i] = f;
}

// block-local exclusive scan of flags (Hillis-Steele in LDS).
// Flags are staged into LDS with the gfx1250 async global->LDS DMA path
// (ASYNCcnt-tracked, bypasses VGPRs), then scanned.
__global__ void k_scan1(const int* __restrict__ flg, int* __restrict__ pre,
                        int* __restrict__ bsum, int n) {
  __shared__ int sh[256];
  int tid = threadIdx.x;
  int i = blockIdx.x * 256 + tid;

  // pre-zero (covers tail lanes that skip the async load)
  sh[tid] = 0;
  __syncthreads();

  if (i < n) {
    uint32_t lds_off = (uint32_t)(uintptr_t)&sh[tid];   // LDS byte offset (addr[31:0])
    const int* gptr = flg + i;
    asm volatile("global_load_async_to_lds_b32 %0, %1, off"
                 :: "v"(lds_off), "v"(gptr) : "memory");
  }
  asm volatile("s_wait_asynccnt 0x0" ::: "memory");
  __syncthreads();

  int v = sh[tid];
  __syncthreads();
  // Hillis-Steele inclusive scan
  for (int off = 1; off < 256; off <<= 1) {
    int t = (tid >= off) ? sh[tid - off] : 0;
    __syncthreads();
    sh[tid] += t;
    __syncthreads();
  }
  if (i < n) pre[i] = sh[tid] - v;            // exclusive within block
  if (tid == 255) bsum[blockIdx.x] = sh[255]; // block total
}

// serial exclusive scan of block sums (few thousand elements); emits voxel_num
__global__ void k_scan2(int* bsum, int nb, float* voxnum_out) {
  if (blockIdx.x == 0 && threadIdx.x == 0) {
    int run = 0;
    for (int b = 0; b < nb; ++b) { int t = bsum[b]; bsum[b] = run; run += t; }
    *voxnum_out = (float)(run < MAX_VOXELS ? run : MAX_VOXELS);
  }
}

// assign ranks (first-occurrence order) into hash values; write coors
__global__ void k_assign(const int* __restrict__ lin, const int* __restrict__ flg,
                         const int* __restrict__ pre, const int* __restrict__ bsum,
                         const int* __restrict__ keys, int* __restrict__ vals,
                         float* __restrict__ coors, int n) {
  int i = blockIdx.x * blockDim.x + threadIdx.x;
  if (i >= n) return;
  if (!flg[i]) return;
  int l = lin[i];
  int r = pre[i] + bsum[i >> 8];
  vals[hfind(keys, l)] = r;
  if (r < MAX_VOXELS) {
    int ix = l / GYZ;
    int rem = l - ix * GYZ;
    int iy = rem / GZ;
    int iz = rem - iy * GZ;
    coors[(size_t)r * 3 + 0] = (float)ix;
    coors[(size_t)r * 3 + 1] = (float)iy;
    coors[(size_t)r * 3 + 2] = (float)iz;
  }
}

// per-point voxel rank (or -1); count points per stored voxel
__global__ void k_vrank(const int* __restrict__ lin, const int* __restrict__ keys,
                        const int* __restrict__ vals, int* __restrict__ vr,
                        int* cnt, int n) {
  int i = blockIdx.x * blockDim.x + threadIdx.x;
  if (i >= n) return;
  int l = lin[i];
  int r = -1;
  if (l >= 0) {
    r = vals[hfind(keys, l)];
    if (r >= MAX_VOXELS) r = -1;
  }
  vr[i] = r;
  if (r >= 0) atomicAdd(&cnt[r], 1);
}

__global__ void k_npv(const int* __restrict__ cnt, float* __restrict__ npv) {
  int r = blockIdx.x * blockDim.x + threadIdx.x;
  if (r < MAX_VOXELS) {
    int c = cnt[r];
    npv[r] = (float)(c < MAX_POINTS ? c : MAX_POINTS);
  }
}

// round A: candidate = min original index strictly greater than threshold
__global__ void k_roundA(const int* __restrict__ vr, const int* __restrict__ thr,
                         int* cmin, int n) {
  int i = blockIdx.x * blockDim.x + threadIdx.x;
  if (i >= n) return;
  __builtin_prefetch(vr + i + 4096, 0, 0);
  int r = vr[i];
  if (r < 0) return;
  if (i > thr[r]) atomicMin(&cmin[r], i);
}

// round B: commit slot k for every voxel that found a candidate
__global__ void k_roundB(const float* __restrict__ pts, int* thr, int* cmin,
                         float* __restrict__ voxels, int k) {
  int r = blockIdx.x * blockDim.x + threadIdx.x;
  if (r >= MAX_VOXELS) return;
  int m = cmin[r];
  if (m == INT_MAX) return;
  thr[r] = m;
  cmin[r] = INT_MAX;
  const float* src = pts + (size_t)m * NFEAT;
  float* dst = voxels + ((size_t)r * MAX_POINTS + k) * NFEAT;
  dst[0] = src[0];
  dst[1] = src[1];
  dst[2] = src[2];
  dst[3] = src[3];
  dst[4] = src[4];
}

// ---------------- host launcher ----------------

extern "C" void kernel_launch(void* const* d_in, const int* in_sizes, int n_in,
                              void* d_out, int out_size, void* d_ws, size_t ws_size,
                              hipStream_t stream) {
  const float* pts = (const float*)d_in[0];
  const int N = in_sizes[0] / NFEAT;          // 1,200,000
  const int NB = (N + 255) / 256;             // scan blocks

  float* out = (float*)d_out;
  float* voxels_o = out;                       // [MAX_VOXELS][10][5]
  float* coors_o  = out + VOX_ELEMS;           // [MAX_VOXELS][3]
  float* npv_o    = out + VOX_ELEMS + COOR_ELEMS;
  float* voxnum_o = out + VOXNUM_OFF;

  // workspace carve-up (ints)
  int* ws   = (int*)d_ws;
  int* keys = ws;                 ws += HCAP;
  int* vals = ws;                 ws += HCAP;
  int* lin  = ws;                 ws += N;
  int* flg  = ws;                 ws += N;
  int* pre  = ws;                 ws += N;
  int* vr   = ws;                 ws += N;
  int* bsum = ws;                 ws += NB + 1;
  int* cnt  = ws;                 ws += MAX_VOXELS;
  int* thr  = ws;                 ws += MAX_VOXELS;
  int* cmin = ws;                 ws += MAX_VOXELS;
  (void)ws_size; (void)n_in;

  const int T = 256;
  const int gN = (N + T - 1) / T;
  const int gV = (MAX_VOXELS + T - 1) / T;
  const int gH = (int)((HCAP + T - 1) / T);
  const int gO = (out_size + T - 1) / T;

  // init
  k_init_hash  <<<gH, T, 0, stream>>>(keys, vals);
  k_init_vstate<<<gV, T, 0, stream>>>(cnt, thr, cmin);
  k_zero_out   <<<gO, T, 0, stream>>>(out, out_size);

  // voxel id per point fused with first-index atomicMin; then flags
  k_lin_first<<<gN, T, 0, stream>>>(pts, lin, keys, vals, N);
  k_flags    <<<gN, T, 0, stream>>>(lin, keys, vals, flg, N);

  // exclusive scan of is_first -> first-occurrence ranks + voxel_num
  k_scan1 <<<NB, T, 0, stream>>>(flg, pre, bsum, N);
  k_scan2 <<<1, 1, 0, stream>>>(bsum, NB, voxnum_o);
  k_assign<<<gN, T, 0, stream>>>(lin, flg, pre, bsum, keys, vals, coors_o, N);

  // per-point rank, per-voxel count, num_points_per_voxel
  k_vrank <<<gN, T, 0, stream>>>(lin, keys, vals, vr, cnt, N);
  k_npv   <<<gV, T, 0, stream>>>(cnt, npv_o);

  // 10 rounds of deterministic k-th-smallest selection -> slot assignment
  for (int k = 0; k < MAX_POINTS; ++k) {
    k_roundA<<<gN, T, 0, stream>>>(vr, thr, cmin, N);
    k_roundB<<<gV, T, 0, stream>>>(pts, thr, cmin, voxels_o, k);
  }
}